// ChebyshevDescriptor_21595095564617
// MI455X (gfx1250) — compile-verified
//
#include <hip/hip_runtime.h>
#include <math.h>

typedef __attribute__((ext_vector_type(2))) float v2f;
typedef __attribute__((ext_vector_type(8))) float v8f;

#define RAD_ORDER 15
#define ANG_ORDER 7
#define RAD_CUTOFF 6.0f
#define ANG_CUTOFF 4.0f
#define MIN_CUTOFF 0.55f
#define MAX_NB 32
#define WAVES_PER_BLOCK 8
#define PI_F 3.14159265358979323846f

// Butterfly sum across a wave32 (all lanes end with the total).
__device__ __forceinline__ float wave_sum32(float v) {
#pragma unroll
  for (int off = 16; off >= 1; off >>= 1) v += __shfl_xor(v, off, 32);
  return v;
}

// One wave32 per atom: lane j owns neighbor j (MAX_NB == warpSize == 32).
// cos(theta) Gram matrix (32x32) = U(32x4-padded) * U^T computed with four
// v_wmma_f32_16x16x4_f32 tiles; Chebyshev recurrences stay in registers.
__global__ __launch_bounds__(256) void cheby_desc_kernel(
    const int* __restrict__ species,   // [N]
    const int* __restrict__ nb_idx,    // [N][32]
    const float* __restrict__ nb_vec,  // [N][32][3]
    float* __restrict__ out,           // [N][48]
    int n_atoms) {
  const int lane = threadIdx.x & 31;
  const int w = threadIdx.x >> 5;
  const int atom = blockIdx.x * WAVES_PER_BLOCK + w;
  const bool active = atom < n_atoms;
  const int a = active ? atom : 0;

  __shared__ float4 sU[WAVES_PER_BLOCK][MAX_NB];   // unit vectors (x,y,z,0)
  __shared__ float sFa[WAVES_PER_BLOCK][MAX_NB];   // angular cutoff weights
  __shared__ float sTs[WAVES_PER_BLOCK][MAX_NB];   // typespin per neighbor

  // ---- per-neighbor loads ----
  const float* vp = nb_vec + ((size_t)a * MAX_NB + lane) * 3;
  float vx = vp[0], vy = vp[1], vz = vp[2];
  int nj = nb_idx[a * MAX_NB + lane];
  float ts = (float)(2 * species[nj] - 1);  // typespin for 2 species: {-1,+1}

  float d = sqrtf(vx * vx + vy * vy + vz * vz);
  float dsafe = fmaxf(d, 1e-8f);

  // ---- radial features (per-lane contribution, reduced at the end) ----
  float maskr = (d <= RAD_CUTOFF && d > MIN_CUTOFF) ? 1.0f : 0.0f;
  float fcr = 0.5f * (__cosf(PI_F * fminf(d, RAD_CUTOFF) / RAD_CUTOFF) + 1.0f);
  float g = fcr * maskr;
  float xr = fminf(fmaxf(2.0f * (d - MIN_CUTOFF) / (RAD_CUTOFF - MIN_CUTOFF) - 1.0f,
                         -1.0f), 1.0f);
  float radu[RAD_ORDER + 1];
  {
    float Tp = 1.0f, Tc = xr;
    radu[0] = g;
    radu[1] = g * xr;
#pragma unroll
    for (int t = 2; t <= RAD_ORDER; ++t) {
      float Tn = 2.0f * xr * Tc - Tp;
      radu[t] = g * Tn;
      Tp = Tc; Tc = Tn;
    }
  }

  // ---- angular prep: unit vector + cutoff weight into LDS ----
  float maska = (d <= ANG_CUTOFF && d > MIN_CUTOFF) ? 1.0f : 0.0f;
  float fa = 0.5f * (__cosf(PI_F * fminf(d, ANG_CUTOFF) / ANG_CUTOFF) + 1.0f) * maska;
  float inv = 1.0f / dsafe;
  sU[w][lane] = make_float4(vx * inv, vy * inv, vz * inv, 0.0f);
  sFa[w][lane] = fa;
  sTs[w][lane] = ts;
  __syncthreads();

  const int half = lane >> 4;   // 0: K=0,1 slots  1: K=2,3 slots
  const int l15 = lane & 15;

  // A(16x4) fragment layout (ISA 7.12.2): lanes 0-15 hold (K0,K1)=(x,y),
  // lanes 16-31 hold (K2,K3)=(z,0). B(4x16) mirrors it, so the same pair
  // serves as both A and B for a given 16-row block of U.
  float4 u0 = sU[w][l15];
  float4 u1 = sU[w][16 + l15];
  v2f P0, P1;
  if (half == 0) {
    P0[0] = u0.x; P0[1] = u0.y;
    P1[0] = u1.x; P1[1] = u1.y;
  } else {
    P0[0] = u0.z; P0[1] = 0.0f;
    P1[0] = u1.z; P1[1] = 0.0f;
  }

  v8f z = {};
  v8f D00 = __builtin_amdgcn_wmma_f32_16x16x4_f32(false, P0, false, P0, (short)0, z, false, false);
  v8f D01 = __builtin_amdgcn_wmma_f32_16x16x4_f32(false, P0, false, P1, (short)0, z, false, false);
  v8f D10 = __builtin_amdgcn_wmma_f32_16x16x4_f32(false, P1, false, P0, (short)0, z, false, false);
  v8f D11 = __builtin_amdgcn_wmma_f32_16x16x4_f32(false, P1, false, P1, (short)0, z, false, false);

  // Per-lane row weights for the 8 rows it owns in each row-block tile.
  float fam[2][8], tsm[2][8];
#pragma unroll
  for (int r = 0; r < 2; ++r) {
    int mb = 16 * r + 8 * half;
#pragma unroll
    for (int p = 0; p < 8; ++p) {
      fam[r][p] = sFa[w][mb + p];
      tsm[r][p] = sTs[w][mb + p];
    }
  }
  float fan[2] = { sFa[w][l15], sFa[w][16 + l15] };
  float tsn[2] = { sTs[w][l15], sTs[w][16 + l15] };

  float angu[ANG_ORDER + 1], angw[ANG_ORDER + 1];
#pragma unroll
  for (int t = 0; t <= ANG_ORDER; ++t) { angu[t] = 0.0f; angw[t] = 0.0f; }

  // C/D layout: VGPR p, lane L -> element (M = p + 8*(L>=16), N = L&15).
#pragma unroll
  for (int r = 0; r < 2; ++r) {
#pragma unroll
    for (int c = 0; c < 2; ++c) {
      v8f Dv = (r == 0) ? (c == 0 ? D00 : D01) : (c == 0 ? D10 : D11);
      int n = 16 * c + l15;
      int mb = 16 * r + 8 * half;
#pragma unroll
      for (int p = 0; p < 8; ++p) {
        int m = mb + p;
        float cv = fminf(fmaxf(Dv[p], -1.0f), 1.0f);
        float wgt = (m < n) ? fam[r][p] * fan[c] : 0.0f;  // strict triu pair weight
        float wtt = wgt * tsm[r][p] * tsn[c];
        float Tp = 1.0f, Tc = cv;
        angu[0] += wgt;      angw[0] += wtt;
        angu[1] += wgt * cv; angw[1] += wtt * cv;
#pragma unroll
        for (int t = 2; t <= ANG_ORDER; ++t) {
          float Tn = 2.0f * cv * Tc - Tp;
          angu[t] += wgt * Tn;
          angw[t] += wtt * Tn;
          Tp = Tc; Tc = Tn;
        }
      }
    }
  }

  // ---- wave-level reductions: [rad_unw(16), rad_w(16), ang_unw(8), ang_w(8)] ----
  float outv[48];
#pragma unroll
  for (int t = 0; t <= RAD_ORDER; ++t) {
    outv[t] = wave_sum32(radu[t]);
    outv[16 + t] = wave_sum32(radu[t] * ts);
  }
#pragma unroll
  for (int t = 0; t <= ANG_ORDER; ++t) {
    outv[32 + t] = wave_sum32(angu[t]);
    outv[40 + t] = wave_sum32(angw[t]);
  }

  if (active && lane == 0) {
    float* o = out + (size_t)atom * 48;
#pragma unroll
    for (int k = 0; k < 48; ++k) o[k] = outv[k];
  }
}

extern "C" void kernel_launch(void* const* d_in, const int* in_sizes, int n_in,
                              void* d_out, int out_size, void* d_ws, size_t ws_size,
                              hipStream_t stream) {
  // setup_inputs order: positions(unused), species_indices, neighbor_indices, neighbor_vectors
  const int* species = (const int*)d_in[1];
  const int* nb_idx = (const int*)d_in[2];
  const float* nb_vec = (const float*)d_in[3];
  float* out = (float*)d_out;
  int n_atoms = in_sizes[1];
  int blocks = (n_atoms + WAVES_PER_BLOCK - 1) / WAVES_PER_BLOCK;
  hipLaunchKernelGGL(cheby_desc_kernel, dim3(blocks), dim3(256), 0, stream,
                     species, nb_idx, nb_vec, out, n_atoms);
}